// get_loss_26938034881138
// MI455X (gfx1250) — compile-verified
//
#include <hip/hip_runtime.h>
#include <hip/hip_bf16.h>

#define PI_F 3.1415926f
#define NB 64
#define NT 8192
#define NPTS 180
#define NPAD 192          // 12 tiles of 16
#define NTILES 12
#define NCHUNKS (NT / 16) // 512
#define WAVES 8
#define ITERS (NCHUNKS / WAVES) // 64, uniform across all waves

typedef __attribute__((ext_vector_type(2))) float v2f;
typedef __attribute__((ext_vector_type(8))) float v8f;

// ---------------------------------------------------------------------------
// Kernel 1: pred [64,70] -> padded points [64,192] as float4 {x,y,z,|p|^2}
// ---------------------------------------------------------------------------
__global__ void circle_points_kernel(const float* __restrict__ pred,
                                     float4* __restrict__ pts) {
    const int b = blockIdx.x;
    const int idx = threadIdx.x; // 0..191
    if (idx >= NPTS) {
        pts[b * NPAD + idx] = make_float4(0.f, 0.f, 0.f, 0.f);
        return;
    }
    const int j = idx / 18;
    const int k = idx % 18;
    const float* p = pred + b * 70 + j * 7;
    const float cx = p[0], cy = p[1], cz = p[2];
    float nx = p[3], ny = p[4], nz = p[5];
    const float r = p[6];

    const float invn = 1.0f / sqrtf(nx * nx + ny * ny + nz * nz);
    nx *= invn; ny *= invn; nz *= invn;

    // cz_vec = normalize([-ny, nx, 0])
    float ax = -ny, ay = nx;
    const float inva = 1.0f / sqrtf(ax * ax + ay * ay);
    ax *= inva; ay *= inva;

    const float fx = cx + r * ax;
    const float fy = cy + r * ay;
    const float fz = cz;

    const float ang = (float)k * (20.0f * 2.0f * PI_F / 360.0f);
    const float C = cosf(ang);
    const float S = sinf(ang);
    const float U = nx, V = ny, W = nz;
    const float omc = 1.0f - C;

    const float r00 = U * U + (1.0f - U * U) * C;
    const float r01 = U * V * omc - W * S;
    const float r02 = U * W * omc + V * S;
    const float r10 = U * V * omc + W * S;
    const float r11 = V * V + (1.0f - V * V) * C;
    const float r12 = V * W * omc - U * S;
    const float r20 = U * W * omc - V * S;
    const float r21 = V * W * omc + U * S;
    const float r22 = W * W + (1.0f - W * W) * C;

    const float x = r00 * fx + r01 * fy + r02 * fz;
    const float y = r10 * fx + r11 * fy + r12 * fz;
    const float z = r20 * fx + r21 * fy + r22 * fz;
    const float p2 = x * x + y * y + z * z;
    pts[b * NPAD + idx] = make_float4(x, y, z, p2);
}

// ---------------------------------------------------------------------------
// Kernel 2: per (batch, 16-point tile) fused WMMA chamfer-min.
//   grid = (12, 64), block = 256 (8 wave32)
//   Uniform trip count (64) so EXEC stays all-1s (WMMA requirement) and the
//   compiler can unroll 4 deep: 4 independent v_wmma in flight per wave.
// ---------------------------------------------------------------------------
__global__ void chamfer_min_kernel(const float* __restrict__ target,
                                   const float4* __restrict__ pts,
                                   float* __restrict__ partials) {
    const int tile = blockIdx.x;
    const int b = blockIdx.y;
    const int lane = threadIdx.x & 31;
    const int wave = threadIdx.x >> 5;
    const int mlane = lane & 15;

    __shared__ float smin[WAVES][16];

    // ---- A fragment: 16 points x K=4 (K=3 real coords, pad 0) ----
    // VGPR0: lanes 0-15 K=0 (x), lanes 16-31 K=2 (z)
    // VGPR1: lanes 0-15 K=1 (y), lanes 16-31 K=3 (0)
    const float4 mypt = pts[b * NPAD + tile * 16 + mlane];
    v2f A;
    if (lane < 16) { A.x = mypt.x; A.y = mypt.y; }
    else           { A.x = mypt.z; A.y = 0.0f;   }

    float rmin[8];
#pragma unroll
    for (int i = 0; i < 8; ++i) rmin[i] = 3.4e38f;

    // this wave's first target element; advances by 8 chunks (1536 B) per iter
    const float* tp0 = target + ((size_t)b * NT + (size_t)wave * 16 + mlane) * 3;

#pragma unroll 4
    for (int it = 0; it < ITERS; ++it) {
        const float* tp = tp0 + (size_t)it * (WAVES * 16 * 3);
        __builtin_prefetch(tp + WAVES * 16 * 3, 0, 1);
        const float tx = tp[0];
        const float ty = tp[1];
        const float tz = tp[2];
        const float t2 = tx * tx + ty * ty + tz * tz;

        // ---- B fragment: K=4 x 16 targets ----
        v2f Bf;
        if (lane < 16) { Bf.x = tx; Bf.y = ty; }
        else           { Bf.x = tz; Bf.y = 0.0f; }

        v8f c = {};
        c = __builtin_amdgcn_wmma_f32_16x16x4_f32(
                /*neg_a=*/false, A, /*neg_b=*/false, Bf,
                /*c_mod=*/(short)0, c, /*reuse_a=*/false, /*reuse_b=*/false);

        // d2 - p2 = t2 - 2*cross ; track running min over n
#pragma unroll
        for (int i = 0; i < 8; ++i) {
            const float cand = t2 - 2.0f * c[i];
            rmin[i] = fminf(rmin[i], cand);
        }
    }

    // min across the 16 n-lanes within each half-wave (xor masks < 16)
#pragma unroll
    for (int off = 1; off < 16; off <<= 1) {
#pragma unroll
        for (int i = 0; i < 8; ++i)
            rmin[i] = fminf(rmin[i], __shfl_xor(rmin[i], off, 32));
    }

    // lane 0 holds mins for m = 0..7, lane 16 for m = 8..15
    if (mlane == 0) {
        const int mbase = (lane >= 16) ? 8 : 0;
#pragma unroll
        for (int i = 0; i < 8; ++i) smin[wave][mbase + i] = rmin[i];
    }
    __syncthreads();

    // wave 0: min across 8 waves, add p2, clamp, mask pads, sum 16 points
    if (threadIdx.x < 32) {
        float contrib = 0.0f;
        if (threadIdx.x < 16) {
            const int m = threadIdx.x;
            float v = smin[0][m];
#pragma unroll
            for (int w = 1; w < WAVES; ++w) v = fminf(v, smin[w][m]);
            const int ptIdx = tile * 16 + m;
            if (ptIdx < NPTS) {
                const float p2 = pts[b * NPAD + ptIdx].w;
                contrib = fmaxf(v + p2, 0.0f);
            }
        }
#pragma unroll
        for (int off = 1; off < 32; off <<= 1)
            contrib += __shfl_xor(contrib, off, 32);
        if (threadIdx.x == 0)
            partials[b * NTILES + tile] = contrib;
    }
}

// ---------------------------------------------------------------------------
// Kernel 3: deterministic reduction of 768 partials -> scalar mean
// ---------------------------------------------------------------------------
__global__ void reduce_kernel(const float* __restrict__ partials,
                              float* __restrict__ out) {
    __shared__ float s[8];
    float v = 0.0f;
    for (int i = threadIdx.x; i < NB * NTILES; i += 256) v += partials[i];
#pragma unroll
    for (int off = 1; off < 32; off <<= 1) v += __shfl_xor(v, off, 32);
    const int wave = threadIdx.x >> 5;
    const int lane = threadIdx.x & 31;
    if (lane == 0) s[wave] = v;
    __syncthreads();
    if (threadIdx.x == 0) {
        float t = 0.0f;
#pragma unroll
        for (int w = 0; w < 8; ++w) t += s[w];
        out[0] = t * (1.0f / ((float)NB * (float)NPTS));
    }
}

// ---------------------------------------------------------------------------
extern "C" void kernel_launch(void* const* d_in, const int* in_sizes, int n_in,
                              void* d_out, int out_size, void* d_ws, size_t ws_size,
                              hipStream_t stream) {
    const float* pred   = (const float*)d_in[0]; // [64,70]
    const float* target = (const float*)d_in[1]; // [64,8192,3]
    // d_in[2] (trans_feat) is unused by the reference loss.
    float* out = (float*)d_out;

    float4* ws_pts  = (float4*)d_ws;                                   // 64*192 float4
    float*  ws_part = (float*)((char*)d_ws + (size_t)NB * NPAD * 16);  // 768 floats

    circle_points_kernel<<<dim3(NB), dim3(NPAD), 0, stream>>>(pred, ws_pts);

    chamfer_min_kernel<<<dim3(NTILES, NB), dim3(256), 0, stream>>>(
        target, ws_pts, ws_part);

    reduce_kernel<<<dim3(1), dim3(256), 0, stream>>>(ws_part, out);
}